// SSIM_1829656068748
// MI455X (gfx1250) — compile-verified
//
#include <hip/hip_runtime.h>

typedef float v2f __attribute__((ext_vector_type(2)));
typedef float v4f __attribute__((ext_vector_type(4)));
typedef float v8f __attribute__((ext_vector_type(8)));

#define IMG_H 512
#define IMG_W 512
#define NBATCH 64
#define WAVES_PER_BLOCK 8

// Staged halo strip: 28 rows x 148 cols (37 b128 groups/row), per image.
#define SROWS 28
#define SSTRIDE 148          /* floats per staged row (= 37 * 4, 16B aligned) */
#define SGRP 37              /* b128 groups per row */
#define SGRP_TOTAL (SROWS * SGRP * 2) /* both images: 2072 groups */

#define MID_STRIDE 34        /* s_mid row stride (even -> b64-aligned pairs) */

__device__ __forceinline__ v8f wmma_f32(v2f a, v2f b, v8f c) {
  // D(16x16,f32) = A(16x4,f32) * B(4x16,f32) + C
  return __builtin_amdgcn_wmma_f32_16x16x4_f32(
      /*neg_a=*/false, a, /*neg_b=*/false, b,
      /*c_mod=*/(short)0, c, /*reuse_a=*/false, /*reuse_b=*/false);
}

// Banded Gaussian weight: g[d] for 0<=d<=10, else 0. Branchless clamped read.
__device__ __forceinline__ float gwin(const float* sg, int d) {
  int dc = d < 0 ? 0 : (d > 10 ? 10 : d);
  float v = sg[dc];
  return (d == dc) ? v : 0.0f;
}

__global__ __launch_bounds__(256) void ssim_tile_kernel(
    const float* __restrict__ img1, const float* __restrict__ img2,
    const float* __restrict__ window, float* __restrict__ partial) {
  __shared__ float s_g[11];
  __shared__ float s_in[2][SROWS * SSTRIDE];               // staged halo strips
  __shared__ float s_mid[WAVES_PER_BLOCK * 16 * MID_STRIDE];
  __shared__ float s_wsum[WAVES_PER_BLOCK];

  // Recover the separable 1-D Gaussian from the 2-D window input:
  // w2d[5][j] = g[5]*g[j], sum_j w2d[5][j] = g[5]  =>  g[j] = w[55+j]/sum.
  if (threadIdx.x < 11) {
    float s = 0.f;
#pragma unroll
    for (int j = 0; j < 11; ++j) s += window[55 + j];
    s_g[threadIdx.x] = window[55 + threadIdx.x] / s;
  }

  // Block -> one 16-row x 128-col output strip (8 tiles, one per wave).
  const int img = blockIdx.x >> 7;        // 128 blocks per image
  const int rem = blockIdx.x & 127;
  const int ty = rem >> 2;                // 32 row-tiles
  const int strip = rem & 3;              // 4 strips of 128 cols
  const int row0 = ty << 4;
  const int cstrip = strip << 7;

  const float* b1 = img1 + (size_t)img * (IMG_H * IMG_W);
  const float* b2 = img2 + (size_t)img * (IMG_H * IMG_W);

  // ---- Cooperative async staging: rows row0-5..row0+22, cols cstrip-8..+139.
  // Clamped addresses are always valid & 16B aligned; a b128 group is either
  // fully inside the image or fully outside (zero-fixed after the wait).
  for (int idx = threadIdx.x; idx < SGRP_TOTAL; idx += 256) {
    const int im = idx >= (SROWS * SGRP);
    const int r2 = idx - im * (SROWS * SGRP);
    const int row = r2 / SGRP;
    const int g = r2 - row * SGRP;
    const int ar = row0 - 5 + row;
    const int ac = cstrip - 8 + (g << 2);
    const int arc = ar < 0 ? 0 : (ar > (IMG_H - 1) ? (IMG_H - 1) : ar);
    const int acc = ac < 0 ? 0 : (ac > (IMG_W - 4) ? (IMG_W - 4) : ac);
    const float* src = (im ? b2 : b1) + arc * IMG_W + acc;
    unsigned long long ga = (unsigned long long)(uintptr_t)src;
    unsigned int ldsoff =
        (unsigned int)(uintptr_t)&s_in[im][row * SSTRIDE + (g << 2)];
    asm volatile("global_load_async_to_lds_b128 %0, %1, off"
                 :
                 : "v"(ldsoff), "v"(ga)
                 : "memory");
  }
  asm volatile("s_wait_asynccnt 0x0" ::: "memory");

  // Zero-fix out-of-image groups (exact SAME zero padding).
  for (int idx = threadIdx.x; idx < SGRP_TOTAL; idx += 256) {
    const int im = idx >= (SROWS * SGRP);
    const int r2 = idx - im * (SROWS * SGRP);
    const int row = r2 / SGRP;
    const int g = r2 - row * SGRP;
    const int ar = row0 - 5 + row;
    const int ac = cstrip - 8 + (g << 2);
    const bool ok = (ar >= 0) && (ar < IMG_H) && (ac >= 0) && (ac <= IMG_W - 4);
    if (!ok) {
      const v4f z = {0.f, 0.f, 0.f, 0.f};
      *(v4f*)&s_in[im][row * SSTRIDE + (g << 2)] = z;
    }
  }
  __syncthreads();

  const int lane = threadIdx.x & 31;
  const int wv = threadIdx.x >> 5;
  const int half = lane >> 4;
  const int l16 = lane & 15;

  // Banded weight chunks: vertical-pass A operand AND (symmetric window)
  // horizontal-pass B operand.
  // A layout 16x4 f32: VGPR0 = (M=l16, K=2*half), VGPR1 = (M=l16, K=2*half+1).
  float wk0[7], wk1[7];
#pragma unroll
  for (int k = 0; k < 7; ++k) {
    int j = 4 * k + 2 * half;
    wk0[k] = gwin(s_g, j - l16);
    wk1[k] = gwin(s_g, j + 1 - l16);
  }

  const v8f vzero = {0.f, 0.f, 0.f, 0.f, 0.f, 0.f, 0.f, 0.f};
  v8f acc[5][2];
#pragma unroll
  for (int q = 0; q < 5; ++q) {
    acc[q][0] = vzero;
    acc[q][1] = vzero;
  }

  // ---- Vertical pass: Mid(16x32) = Gv(16x28) x In(28x32), 7 K-chunks of 4.
  // B layout 4x16 f32 (mirrors A): VGPR0 = (K=2*half, N=l16), VGPR1 = K+1.
  // Staged-strip column for this wave's tile: sc = wv*16 + 3 + cext.
#pragma unroll
  for (int cc = 0; cc < 2; ++cc) {
    const int sc = wv * 16 + 3 + cc * 16 + l16;
#pragma unroll
    for (int k = 0; k < 7; ++k) {
      const int j = 4 * k + 2 * half;
      const float x1a = s_in[0][j * SSTRIDE + sc];
      const float x1b = s_in[0][(j + 1) * SSTRIDE + sc];
      const float x2a = s_in[1][j * SSTRIDE + sc];
      const float x2b = s_in[1][(j + 1) * SSTRIDE + sc];
      v2f a = {wk0[k], wk1[k]};
      v2f d0 = {x1a, x1b};
      v2f d1 = {x2a, x2b};
      v2f d2 = {x1a * x1a, x1b * x1b};
      v2f d3 = {x2a * x2a, x2b * x2b};
      v2f d4 = {x1a * x2a, x1b * x2b};
      acc[0][cc] = wmma_f32(a, d0, acc[0][cc]);
      acc[1][cc] = wmma_f32(a, d1, acc[1][cc]);
      acc[2][cc] = wmma_f32(a, d2, acc[2][cc]);
      acc[3][cc] = wmma_f32(a, d3, acc[3][cc]);
      acc[4][cc] = wmma_f32(a, d4, acc[4][cc]);
    }
  }

  // ---- Horizontal pass per quantity via this wave's private LDS slice.
  // C/D layout -> A layout is a cross-lane transpose; LDS round-trip.
  float* smid = &s_mid[wv * (16 * MID_STRIDE)];
  v8f outq[5];
#pragma unroll
  for (int q = 0; q < 5; ++q) {
    // scatter D-layout accumulator: elem r of lane = Mid[r+8*half][cc*16+l16]
#pragma unroll
    for (int cc = 0; cc < 2; ++cc) {
#pragma unroll
      for (int r = 0; r < 8; ++r) {
        smid[(r + 8 * half) * MID_STRIDE + cc * 16 + l16] = acc[q][cc][r];
      }
    }
    // same-wave LDS ops are in order; loads below see the stores above
    v8f o = vzero;
#pragma unroll
    for (int k = 0; k < 7; ++k) {
      const int cx = 4 * k + 2 * half;  // even -> 8B aligned pair
      v2f a = *(const v2f*)(smid + l16 * MID_STRIDE + cx);  // Mid as A operand
      v2f b = {wk0[k], wk1[k]};  // Gh chunk == Gv chunk (symmetric window)
      o = wmma_f32(a, b, o);
    }
    outq[q] = o;
  }

  // ---- SSIM map + hierarchical reduction (deterministic, no float atomics)
  const float C1c = 0.01f * 0.01f;
  const float C2c = 0.03f * 0.03f;
  float ssum = 0.f;
#pragma unroll
  for (int e = 0; e < 8; ++e) {
    float mu1 = outq[0][e], mu2 = outq[1][e];
    float mu1sq = mu1 * mu1, mu2sq = mu2 * mu2, mu12 = mu1 * mu2;
    float s11 = outq[2][e] - mu1sq;
    float s22 = outq[3][e] - mu2sq;
    float s12 = outq[4][e] - mu12;
    float num = (2.f * mu12 + C1c) * (2.f * s12 + C2c);
    float den = (mu1sq + mu2sq + C1c) * (s11 + s22 + C2c);
    ssum += num / den;
  }
#pragma unroll
  for (int off = 16; off > 0; off >>= 1) ssum += __shfl_xor(ssum, off, 32);
  if (lane == 0) s_wsum[wv] = ssum;
  __syncthreads();
  if (threadIdx.x == 0) {
    float s = 0.f;
#pragma unroll
    for (int w = 0; w < WAVES_PER_BLOCK; ++w) s += s_wsum[w];
    partial[blockIdx.x] = s;
  }
}

__global__ __launch_bounds__(256) void ssim_reduce_kernel(
    const float* __restrict__ partial, float* __restrict__ out, int n,
    float scale) {
  __shared__ float sh[256];
  float s = 0.f;
  for (int i = threadIdx.x; i < n; i += 256) s += partial[i];
  sh[threadIdx.x] = s;
  __syncthreads();
  for (int off = 128; off > 0; off >>= 1) {
    if ((int)threadIdx.x < off) sh[threadIdx.x] += sh[threadIdx.x + off];
    __syncthreads();
  }
  if (threadIdx.x == 0) out[0] = sh[0] * scale;
}

extern "C" void kernel_launch(void* const* d_in, const int* in_sizes, int n_in,
                              void* d_out, int out_size, void* d_ws,
                              size_t ws_size, hipStream_t stream) {
  const float* img1 = (const float*)d_in[0];
  const float* img2 = (const float*)d_in[1];
  const float* window = (const float*)d_in[2];
  float* out = (float*)d_out;
  float* partial = (float*)d_ws;  // 8192 floats, fully rewritten every call

  const int nblocks = NBATCH * 32 * 4;  // 8192 strips of 16x128
  ssim_tile_kernel<<<nblocks, 256, 0, stream>>>(img1, img2, window, partial);

  const float scale = 1.0f / (float)((size_t)NBATCH * IMG_H * IMG_W);
  ssim_reduce_kernel<<<1, 256, 0, stream>>>(partial, out, nblocks, scale);
}